// GPR_prop_attention_57191784513882
// MI455X (gfx1250) — compile-verified
//
#include <hip/hip_runtime.h>

typedef float v2f __attribute__((ext_vector_type(2)));
typedef float v8f __attribute__((ext_vector_type(8)));

#define LEAKY_SLOPE 0.01f
#define ORD_NEG_INF 0x007FFFFFu  // ordered encoding of -inf

__device__ __forceinline__ unsigned f2ord(float f) {
    unsigned u = __float_as_uint(f);
    return (u & 0x80000000u) ? ~u : (u | 0x80000000u);
}
__device__ __forceinline__ float ord2f(unsigned u) {
    return (u & 0x80000000u) ? __uint_as_float(u & 0x7fffffffu)
                             : __uint_as_float(~u);
}

// -------------------------------------------------------------------------
// Fused: s1 = H @ w_src, s2 = H @ w_dst via V_WMMA_F32_16X16X4_F32 chains,
// plus th (+)= coef * H reusing the preloaded A fragments.
// One wave32 per 16-node tile; 16 chained K=4 WMMAs cover the D=64 reduction.
// A-frag (16x4 f32): lane M = lane&15, K-pair = (lane<16) ? {0,1} : {2,3}.
// B-frag (4x16 f32): lane N = lane&15, same K-pair. Only cols 0,1 nonzero.
// D (16x16 f32): vgpr v, lanes0-15 -> M=v, N=lane; lanes16-31 -> M=v+8.
// Branchless B construction + template ACC: no EXEC divergence in hot loop.
// -------------------------------------------------------------------------
template <int ACC>
__global__ __launch_bounds__(256) void dots_wmma_kernel(
    const float* __restrict__ h, const float* __restrict__ attn_w,
    const float* __restrict__ temp, int temp_idx, float* __restrict__ th,
    float* __restrict__ s1, float* __restrict__ s2, int N) {
    int wave = blockIdx.x * (blockDim.x >> 5) + (threadIdx.x >> 5);
    int base = wave * 16;
    if (base >= N) return;  // wave-uniform: EXEC stays all-1s for WMMA

    int lane = threadIdx.x & 31;
    int M    = lane & 15;
    int kh   = lane >> 4;  // float2-pair selector: 0 -> K={0,1}, 1 -> K={2,3}
    int row  = base + M;
    bool row_valid = row < N;
    int safe_row = row_valid ? row : (N - 1);

    const v2f* hrow2 = (const v2f*)(h + (size_t)safe_row * 64);
    v2f*       trow2 = (v2f*)(th + (size_t)safe_row * 64);
    const v2f* w2    = (const v2f*)attn_w;  // [0..31]=w_src pairs, [32..63]=w_dst
    float coef = temp[temp_idx];

    // Branchless column masks: col 0 = w_src, col 1 = w_dst, rest zero.
    float msrc = (M == 0) ? 1.f : 0.f;
    float mdst = (M == 1) ? 1.f : 0.f;

    // Preload all fragments (no divergence, pure b64 loads).
    v2f a[16], b[16];
#pragma unroll
    for (int c = 0; c < 16; ++c) {
        int idx = 2 * c + kh;
        a[c] = hrow2[idx];
        b[c] = msrc * w2[idx] + mdst * w2[32 + idx];
    }

    // Fused AXPY on th, reusing the A fragments.
    if (row_valid) {
#pragma unroll
        for (int c = 0; c < 16; ++c) {
            int idx = 2 * c + kh;
            if (ACC) {
                v2f t = trow2[idx];
                t[0] += coef * a[c][0];
                t[1] += coef * a[c][1];
                trow2[idx] = t;
            } else {
                v2f t;
                t[0] = coef * a[c][0];
                t[1] = coef * a[c][1];
                trow2[idx] = t;
            }
        }
    }

    // Back-to-back WMMA chain (D -> SRC2 accumulation path).
    v8f acc = {0.f, 0.f, 0.f, 0.f, 0.f, 0.f, 0.f, 0.f};
#pragma unroll
    for (int c = 0; c < 16; ++c) {
        acc = __builtin_amdgcn_wmma_f32_16x16x4_f32(
            false, a[c], false, b[c], (short)0, acc, false, false);
    }

    // Extract: D column 0 -> s1, column 1 -> s2.
    int half = kh * 8;
#pragma unroll
    for (int v = 0; v < 8; ++v) {
        int n = base + half + v;
        if (n < N) {
            if (M == 0) s1[n] = acc[v];
            else if (M == 1) s2[n] = acc[v];
        }
    }
}

// Zero h_next, reset segment-max (ordered -inf) and segment-sum.
__global__ void init_kernel(float* __restrict__ hnext,
                            unsigned* __restrict__ mord,
                            float* __restrict__ ssum, int N, int ND) {
    int i = blockIdx.x * blockDim.x + threadIdx.x;
    if (i < ND) hnext[i] = 0.f;
    if (i < N) {
        mord[i] = ORD_NEG_INF;
        ssum[i] = 0.f;
    }
}

// Per-edge: e = leaky_relu(s1[src]+s2[dst]); segment max via ordered-uint atomicMax.
__global__ void edge_max_kernel(const float* __restrict__ s1,
                                const float* __restrict__ s2,
                                const int* __restrict__ src,
                                const int* __restrict__ dst,
                                float* __restrict__ ebuf,
                                unsigned* __restrict__ mord, int E) {
    int i = blockIdx.x * blockDim.x + threadIdx.x;
    if (i >= E) return;
    float e = s1[src[i]] + s2[dst[i]];
    e = e > 0.f ? e : LEAKY_SLOPE * e;
    ebuf[i] = e;
    atomicMax(&mord[dst[i]], f2ord(e));
}

// Per-edge: a = exp(e - m[dst]); segment sum via f32 atomicAdd.
__global__ void edge_exp_kernel(const float* __restrict__ ebuf,
                                const unsigned* __restrict__ mord,
                                const int* __restrict__ dst,
                                float* __restrict__ abuf,
                                float* __restrict__ ssum, int E) {
    int i = blockIdx.x * blockDim.x + threadIdx.x;
    if (i >= E) return;
    int d = dst[i];
    float a = __expf(ebuf[i] - ord2f(mord[d]));
    abuf[i] = a;
    atomicAdd(&ssum[d], a);
}

// 16 lanes per edge, float4 gather of h[src], 4 atomicAdds into h_next[dst].
__global__ void aggregate_kernel(const float* __restrict__ hprev,
                                 const float* __restrict__ abuf,
                                 const float* __restrict__ ssum,
                                 const int* __restrict__ src,
                                 const int* __restrict__ dst,
                                 float* __restrict__ hnext, int E) {
    int gid = blockIdx.x * blockDim.x + threadIdx.x;
    int e = gid >> 4;
    if (e >= E) return;
    int sub = gid & 15;
    int s = src[e], d = dst[e];
    float w = abuf[e] / ssum[d];
    const float4 v = ((const float4*)(hprev + (size_t)s * 64))[sub];
    float* o = hnext + (size_t)d * 64 + sub * 4;
    atomicAdd(o + 0, v.x * w);
    atomicAdd(o + 1, v.y * w);
    atomicAdd(o + 2, v.z * w);
    atomicAdd(o + 3, v.w * w);
}

// Final th += temp[K] * h_K.
__global__ void axpy_kernel(float* __restrict__ th, const float* __restrict__ h,
                            const float* __restrict__ temp, int temp_idx,
                            int ND) {
    int i = blockIdx.x * blockDim.x + threadIdx.x;
    if (i < ND) th[i] += temp[temp_idx] * h[i];
}

extern "C" void kernel_launch(void* const* d_in, const int* in_sizes, int n_in,
                              void* d_out, int out_size, void* d_ws,
                              size_t ws_size, hipStream_t stream) {
    const float* x      = (const float*)d_in[0];
    const float* attn_w = (const float*)d_in[1];
    const float* temp   = (const float*)d_in[2];
    const int*   src    = (const int*)d_in[3];
    const int*   dst    = (const int*)d_in[4];

    const int D  = 64;
    const int N  = in_sizes[0] / D;
    const int E  = in_sizes[3];
    const int K  = in_sizes[2] - 1;
    const int ND = N * D;

    float* th = (float*)d_out;

    // Workspace carve-up.
    float*    hA   = (float*)d_ws;
    float*    hB   = hA + ND;
    float*    s1   = hB + ND;
    float*    s2   = s1 + N;
    unsigned* mord = (unsigned*)(s2 + N);
    float*    ssum = (float*)(mord + N);
    float*    ebuf = ssum + N;
    float*    abuf = ebuf + E;

    const int BLK         = 256;
    const int tiles       = (N + 15) / 16;
    const int dots_blocks = (tiles + 7) / 8;  // 8 waves / block
    const int nd_blocks   = (ND + BLK - 1) / BLK;
    const int e_blocks    = (E + BLK - 1) / BLK;
    const int agg_blocks  = (E * 16 + BLK - 1) / BLK;

    const float* hprev = x;
    for (int k = 1; k <= K; ++k) {
        float* hnext = (k & 1) ? hA : hB;
        // s1/s2 via WMMA; fused th (+)= temp[k-1] * h_{k-1}
        if (k == 1) {
            dots_wmma_kernel<0><<<dots_blocks, BLK, 0, stream>>>(
                hprev, attn_w, temp, k - 1, th, s1, s2, N);
        } else {
            dots_wmma_kernel<1><<<dots_blocks, BLK, 0, stream>>>(
                hprev, attn_w, temp, k - 1, th, s1, s2, N);
        }
        init_kernel<<<nd_blocks, BLK, 0, stream>>>(hnext, mord, ssum, N, ND);
        edge_max_kernel<<<e_blocks, BLK, 0, stream>>>(s1, s2, src, dst, ebuf,
                                                      mord, E);
        edge_exp_kernel<<<e_blocks, BLK, 0, stream>>>(ebuf, mord, dst, abuf,
                                                      ssum, E);
        aggregate_kernel<<<agg_blocks, BLK, 0, stream>>>(hprev, abuf, ssum, src,
                                                         dst, hnext, E);
        hprev = hnext;
    }
    axpy_kernel<<<nd_blocks, BLK, 0, stream>>>(th, hprev, temp, K, ND);
}